// Decoder_49297634623542
// MI455X (gfx1250) — compile-verified
//
#include <hip/hip_runtime.h>

#define H   2048
#define T   1024
#define OD  512
#define NWG 256
#define NTH 256                      // 8 wave32s per workgroup
#define WPW (NTH / 32)               // waves per workgroup
#define TOTAL_WAVES (NWG * WPW)      // 2048 == H (one wave per h element in phase A)

typedef float v2f __attribute__((ext_vector_type(2)));
typedef float v8f __attribute__((ext_vector_type(8)));

// ---------------------------------------------------------------------------
// Device-wide sense-reversing barrier in global scratch.
// bar[0] = arrival counter, bar[1] = generation.
// ---------------------------------------------------------------------------
__device__ __forceinline__ void grid_sync(unsigned* bar) {
    __syncthreads();
    __threadfence();   // publish our stores (global_wb) before arriving
    if (threadIdx.x == 0) {
        unsigned gen = __hip_atomic_load(&bar[1], __ATOMIC_ACQUIRE,
                                         __HIP_MEMORY_SCOPE_AGENT);
        unsigned prev = __hip_atomic_fetch_add(&bar[0], 1u, __ATOMIC_ACQ_REL,
                                               __HIP_MEMORY_SCOPE_AGENT);
        if (prev == (unsigned)(NWG - 1)) {
            __hip_atomic_store(&bar[0], 0u, __ATOMIC_RELAXED,
                               __HIP_MEMORY_SCOPE_AGENT);
            __hip_atomic_fetch_add(&bar[1], 1u, __ATOMIC_RELEASE,
                                   __HIP_MEMORY_SCOPE_AGENT);
        } else {
            while (__hip_atomic_load(&bar[1], __ATOMIC_ACQUIRE,
                                     __HIP_MEMORY_SCOPE_AGENT) == gen) {
                __builtin_amdgcn_s_sleep(2);
            }
        }
    }
    __syncthreads();
    __threadfence();   // acquire side: invalidate near caches on every CU
}

__global__ void init_barrier_kernel(unsigned* bar) {
    bar[0] = 0u;
    bar[1] = 0u;
}

// ---------------------------------------------------------------------------
// One 16x16 output tile of D = A(1024xH) * W(NxH)^T + bias, via
// V_WMMA_F32_16X16X4_F32.  A fragment: 16x4 (M x K); B fragment: 4x16 (K x N)
// where B[k][n] = W[j0+n][k] -> same per-lane addressing as A on row-major W.
// ---------------------------------------------------------------------------
__device__ __forceinline__ void gemm_all(const float* __restrict__ A,
                                         const float* __restrict__ W,
                                         const float* __restrict__ bias,
                                         float* __restrict__ D, int N) {
    const int lane = threadIdx.x & 31;
    const int wid  = blockIdx.x * WPW + (threadIdx.x >> 5);
    const int half = lane >> 4;      // selects K pair (ISA 16x4 f32 A layout)
    const int lrow = lane & 15;      // M row for A, N col for B
    const int tiles_n = N >> 4;
    const int ntiles  = (T >> 4) * tiles_n;

    for (int tile = wid; tile < ntiles; tile += TOTAL_WAVES) {
        const int tm = tile / tiles_n;
        const int tn = tile - tm * tiles_n;
        const int t0 = tm << 4;
        const int j0 = tn << 4;

        const float* arow = A + (size_t)(t0 + lrow) * H + 2 * half;
        const float* wrow = W + (size_t)(j0 + lrow) * H + 2 * half;

        v8f acc = {};
        for (int k = 0; k < H; k += 4) {
            v2f a = *(const v2f*)(arow + k);   // A[m][2h], A[m][2h+1]
            v2f b = *(const v2f*)(wrow + k);   // W[n][2h], W[n][2h+1]
            acc = __builtin_amdgcn_wmma_f32_16x16x4_f32(
                false, a, false, b, (short)0, acc, false, false);
        }

        const float bj = bias[j0 + lrow];
#pragma unroll
        for (int r = 0; r < 8; ++r) {
            const int m = r + 8 * half;        // C/D layout: M = vgpr + 8*half
            D[(size_t)(t0 + m) * N + (j0 + lrow)] = acc[r] + bj;
        }
    }
}

// ---------------------------------------------------------------------------
// Persistent kernel: Phase A = sequential GRU recurrence (one wave per h[i]),
// Phase B = dense chain as WMMA GEMMs over all 1024 timesteps at once.
// ---------------------------------------------------------------------------
__global__ __launch_bounds__(NTH) void decoder_persistent(
    const float* __restrict__ hidden,
    const float* __restrict__ Wih, const float* __restrict__ Whh,
    const float* __restrict__ bih, const float* __restrict__ bhh,
    const float* __restrict__ W1, const float* __restrict__ b1,
    const float* __restrict__ W2, const float* __restrict__ b2,
    const float* __restrict__ W3, const float* __restrict__ b3,
    const float* __restrict__ W4, const float* __restrict__ b4,
    const float* __restrict__ Wo, const float* __restrict__ bo,
    float* __restrict__ out,
    float* __restrict__ Hs, float* __restrict__ P0, float* __restrict__ P1,
    unsigned* __restrict__ bar) {
    const int lane = threadIdx.x & 31;
    const int i    = blockIdx.x * WPW + (threadIdx.x >> 5);  // h index, 0..H-1

    // -------- Phase A: h_{t} = GRUCell(x_t, h_{t-1}),  x_t = h_{t-1} (t>0) --
    const float* wr_i = Wih + (size_t)i * H;
    const float* wz_i = Wih + (size_t)(H + i) * H;
    const float* wn_i = Wih + (size_t)(2 * H + i) * H;
    const float* wr_h = Whh + (size_t)i * H;
    const float* wz_h = Whh + (size_t)(H + i) * H;
    const float* wn_h = Whh + (size_t)(2 * H + i) * H;

    for (int t = 0; t < T; ++t) {
        const float* hprev = (t == 0) ? hidden : (Hs + (size_t)(t - 1) * H);
        const bool xzero = (t == 0);   // first step: x = 0 vector

        float sir = 0.f, siz = 0.f, sin_ = 0.f;
        float shr = 0.f, shz = 0.f, shn = 0.f;
        for (int k = lane * 4; k < H; k += 128) {
            const float4 hv = *(const float4*)(hprev + k);
            const float4 xv = xzero ? make_float4(0.f, 0.f, 0.f, 0.f) : hv;
            const float4 a = *(const float4*)(wr_i + k);
            const float4 b = *(const float4*)(wz_i + k);
            const float4 c = *(const float4*)(wn_i + k);
            const float4 d = *(const float4*)(wr_h + k);
            const float4 e = *(const float4*)(wz_h + k);
            const float4 f = *(const float4*)(wn_h + k);
            sir  = fmaf(a.x, xv.x, fmaf(a.y, xv.y, fmaf(a.z, xv.z, fmaf(a.w, xv.w, sir))));
            siz  = fmaf(b.x, xv.x, fmaf(b.y, xv.y, fmaf(b.z, xv.z, fmaf(b.w, xv.w, siz))));
            sin_ = fmaf(c.x, xv.x, fmaf(c.y, xv.y, fmaf(c.z, xv.z, fmaf(c.w, xv.w, sin_))));
            shr  = fmaf(d.x, hv.x, fmaf(d.y, hv.y, fmaf(d.z, hv.z, fmaf(d.w, hv.w, shr))));
            shz  = fmaf(e.x, hv.x, fmaf(e.y, hv.y, fmaf(e.z, hv.z, fmaf(e.w, hv.w, shz))));
            shn  = fmaf(f.x, hv.x, fmaf(f.y, hv.y, fmaf(f.z, hv.z, fmaf(f.w, hv.w, shn))));
        }
#pragma unroll
        for (int off = 16; off > 0; off >>= 1) {
            sir  += __shfl_down(sir, off, 32);
            siz  += __shfl_down(siz, off, 32);
            sin_ += __shfl_down(sin_, off, 32);
            shr  += __shfl_down(shr, off, 32);
            shz  += __shfl_down(shz, off, 32);
            shn  += __shfl_down(shn, off, 32);
        }
        if (lane == 0) {
            const float r = 1.0f / (1.0f + __expf(-(sir + bih[i] + shr + bhh[i])));
            const float z = 1.0f / (1.0f + __expf(-(siz + bih[H + i] + shz + bhh[H + i])));
            const float n = tanhf(sin_ + bih[2 * H + i] + r * (shn + bhh[2 * H + i]));
            Hs[(size_t)t * H + i] = (1.0f - z) * n + z * hprev[i];
        }
        grid_sync(bar);   // full h_t visible to everyone before step t+1
    }

    // -------- Phase B: ys = ((((Hs W1^T) W2^T) W3^T) W4^T) Wo^T  (WMMA) -----
    gemm_all(Hs, W1, b1, P0, H);  grid_sync(bar);
    gemm_all(P0, W2, b2, P1, H);  grid_sync(bar);
    gemm_all(P1, W3, b3, P0, H);  grid_sync(bar);
    gemm_all(P0, W4, b4, P1, H);  grid_sync(bar);
    gemm_all(P1, Wo, bo, out, OD);
}

// ---------------------------------------------------------------------------
// Host launcher.  Scratch layout (floats): Hs[T*H] | P0[T*H] | P1[T*H] | bar.
// Requires ws_size >= 24 MB + 8 B.
// ---------------------------------------------------------------------------
extern "C" void kernel_launch(void* const* d_in, const int* in_sizes, int n_in,
                              void* d_out, int out_size, void* d_ws, size_t ws_size,
                              hipStream_t stream) {
    const float* hidden = (const float*)d_in[0];
    // d_in[1] = target: only its length (T) matters; values unused.
    const float* Wih = (const float*)d_in[2];
    const float* Whh = (const float*)d_in[3];
    const float* bih = (const float*)d_in[4];
    const float* bhh = (const float*)d_in[5];
    const float* W1  = (const float*)d_in[6];
    const float* b1  = (const float*)d_in[7];
    const float* W2  = (const float*)d_in[8];
    const float* b2  = (const float*)d_in[9];
    const float* W3  = (const float*)d_in[10];
    const float* b3  = (const float*)d_in[11];
    const float* W4  = (const float*)d_in[12];
    const float* b4  = (const float*)d_in[13];
    const float* Wo  = (const float*)d_in[14];
    const float* bo  = (const float*)d_in[15];

    float* out = (float*)d_out;
    float* Hs  = (float*)d_ws;
    float* P0  = Hs + (size_t)T * H;
    float* P1  = P0 + (size_t)T * H;
    unsigned* bar = (unsigned*)(P1 + (size_t)T * H);

    init_barrier_kernel<<<1, 1, 0, stream>>>(bar);
    decoder_persistent<<<NWG, NTH, 0, stream>>>(
        hidden, Wih, Whh, bih, bhh,
        W1, b1, W2, b2, W3, b3, W4, b4, Wo, bo,
        out, Hs, P0, P1, bar);
}